// TrackHead_22187801051266
// MI455X (gfx1250) — compile-verified
//
#include <hip/hip_runtime.h>

typedef __attribute__((ext_vector_type(2))) float v2f;
typedef __attribute__((ext_vector_type(4))) float v4f;
typedef __attribute__((ext_vector_type(8))) float v8f;

#define BM 128
#define BN 64
#define BK 32
#define BKP 36   // padded k-stride (words): even (b64-aligned), mult of 4 (b128-aligned),
                 // 36*l16 mod 64 distinct for l16=0..15 -> conflict-free b64 reads

// C[M x N] = optRelu(A[M x K] @ B + bias), A row-major (lda).
// transB==0: B row-major [K x N] (ldb). transB==1: B is [N x K] row-major, used as B^T.
// V_WMMA_F32_16X16X4_F32; 256 threads = 8 waves; block tile 128x64;
// each wave: 16 rows x 4 column tiles. B staged in LDS k-contiguous per column so each
// B fragment is a single aligned ds_load_b64 into the WMMA operand pair.
__global__ __launch_bounds__(256)
void gemm_wmma(const float* __restrict__ A, const float* __restrict__ B,
               const float* __restrict__ bias, float* __restrict__ C,
               int M, int N, int K, int lda, int ldb, int ldc,
               int transB, int relu)
{
    __shared__ float Bs[BN][BKP];   // [n][k]

    const int tid  = threadIdx.x;
    const int wave = tid >> 5;
    const int lane = tid & 31;
    const int half = lane >> 4;   // 0: lanes 0-15, 1: lanes 16-31
    const int l16  = lane & 15;

    const int rowBlock = blockIdx.y * BM;
    const int colBlock = blockIdx.x * BN;
    const int waveRow  = rowBlock + wave * 16;

    // A-fragment row for this lane (clamped so partial tiles keep EXEC all-1s)
    int am = waveRow + l16;
    if (am > M - 1) am = M - 1;
    const float* aRow = A + (size_t)am * lda;

    v8f acc[BN / 16] = {};

    for (int k0 = 0; k0 < K; k0 += BK) {
        __syncthreads();
        if (!transB) {
            // Bs[n][k] = B[(k0+k)*ldb + colBlock+n]; 2 reps x (256 thr x 4 floats) = 32x64
#pragma unroll
            for (int rep = 0; rep < 2; ++rep) {
                int idx = tid * 4 + rep * 1024;
                int k = idx >> 6;         // 0..31
                int n = idx & 63;
                const float* gp = B + (size_t)(k0 + k) * ldb + (colBlock + n);
                v4f v = *(const v4f*)gp;
                Bs[n + 0][k] = v.x; Bs[n + 1][k] = v.y;
                Bs[n + 2][k] = v.z; Bs[n + 3][k] = v.w;
            }
        } else {
            // Bs[n][k] = B[(colBlock+n)*ldb + k0+k]; k-contiguous both sides
            int n  = tid >> 2;            // 0..63
            int kb = (tid & 3) * 8;       // 0,8,16,24
            int rn = colBlock + n;
            if (rn > N - 1) rn = N - 1;
            const float* gp = B + (size_t)rn * ldb + (k0 + kb);
            *(v4f*)&Bs[n][kb + 0] = *(const v4f*)(gp + 0);
            *(v4f*)&Bs[n][kb + 4] = *(const v4f*)(gp + 4);
        }
        __syncthreads();

        if (k0 + BK < K)  // prefetch this lane's next A chunk (global_prefetch_b8)
            __builtin_prefetch(aRow + k0 + BK, 0, 0);

#pragma unroll
        for (int kk = 0; kk < BK; kk += 4) {
            // 16x4 fp32 A fragment: v0 = K 0 (lanes 0-15) / K 2 (lanes 16-31), v1 = K 1 / K 3
            v2f a = *(const v2f*)(aRow + k0 + kk + half * 2);
#pragma unroll
            for (int t = 0; t < BN / 16; ++t) {
                v2f b = *(const v2f*)&Bs[t * 16 + l16][kk + half * 2];
                acc[t] = __builtin_amdgcn_wmma_f32_16x16x4_f32(
                    false, a, false, b, (short)0, acc[t], false, false);
            }
        }
    }

    // Epilogue: C/D layout — VGPR i: lanes 0-15 -> row i, lanes 16-31 -> row 8+i
#pragma unroll
    for (int t = 0; t < BN / 16; ++t) {
        int col = colBlock + t * 16 + l16;
        if (col < N) {
            float bv = bias ? bias[col] : 0.0f;
#pragma unroll
            for (int i = 0; i < 8; ++i) {
                int row = waveRow + half * 8 + i;
                if (row < M) {
                    float v = acc[t][i] + bv;
                    if (relu) v = fmaxf(v, 0.0f);
                    C[(size_t)row * ldc + col] = v;
                }
            }
        }
    }
}

__global__ void iou_kernel(const float* __restrict__ bb, const float* __restrict__ rbb,
                           float* __restrict__ ious, float* __restrict__ dxp,
                           float* __restrict__ dyp, int N, int M, int ldc)
{
    int j = blockIdx.x * blockDim.x + threadIdx.x;   // ref box index
    int i = blockIdx.y;                              // box index
    if (j >= M) return;
    float bx1 = bb[i*4+0], by1 = bb[i*4+1], bx2 = bb[i*4+2], by2 = bb[i*4+3];
    float rx1 = rbb[j*4+0], ry1 = rbb[j*4+1], rx2 = rbb[j*4+2], ry2 = rbb[j*4+3];
    float cx  = (bx1 + bx2) * 0.5f, cy  = (by1 + by2) * 0.5f;
    float rcx = (rx1 + rx2) * 0.5f, rcy = (ry1 + ry2) * 0.5f;
    float dx = rcx - cx, dy = rcy - cy;
    float sx1 = bx1 + dx, sy1 = by1 + dy, sx2 = bx2 + dx, sy2 = by2 + dy;
    float iw = fmaxf(fminf(sx2, rx2) - fmaxf(sx1, rx1) + 1.0f, 0.0f);
    float ih = fmaxf(fminf(sy2, ry2) - fmaxf(sy1, ry1) + 1.0f, 0.0f);
    float inter = iw * ih;
    float as_ = (sx2 - sx1 + 1.0f) * (sy2 - sy1 + 1.0f);
    float ar  = (rx2 - rx1 + 1.0f) * (ry2 - ry1 + 1.0f);
    float iou = inter / (as_ + ar - inter);
    size_t o = (size_t)i * ldc + j + 1;
    ious[o] = iou; dxp[o] = dx; dyp[o] = dy;
}

__global__ void edge_kernel(float* __restrict__ match, float* __restrict__ shift,
                            float* __restrict__ ious, float* __restrict__ dxp,
                            float* __restrict__ dyp, int N, int ldc)
{
    int i = blockIdx.x * blockDim.x + threadIdx.x;
    if (i < N) {
        match[(size_t)i * ldc] = 0.0f;
        ious [(size_t)i * ldc] = 0.0f;
        dxp  [(size_t)i * ldc] = 0.0f;
        dyp  [(size_t)i * ldc] = 0.0f;
    }
    if (i == 0) *shift = 0.0f;
}

extern "C" void kernel_launch(void* const* d_in, const int* in_sizes, int n_in,
                              void* d_out, int out_size, void* d_ws, size_t ws_size,
                              hipStream_t stream)
{
    (void)n_in; (void)out_size; (void)ws_size;
    const float* bboxes  = (const float*)d_in[0];
    const float* rbboxes = (const float*)d_in[1];
    const float* x       = (const float*)d_in[2];
    const float* rx      = (const float*)d_in[3];
    const float* W0      = (const float*)d_in[4];
    const float* b0      = (const float*)d_in[5];
    const float* W1      = (const float*)d_in[6];
    const float* b1      = (const float*)d_in[7];

    const int N   = in_sizes[0] / 4;      // 2000
    const int M   = in_sizes[1] / 4;      // 2000
    const int D   = in_sizes[5];          // 1024
    const int Fin = in_sizes[4] / D;      // 12544
    const int ldc = M + 1;                // 2001

    // Workspace: HH[(N+M) x D], G[(N+M) x D]  (~32 MB)
    float* HH = (float*)d_ws;
    float* G  = HH + (size_t)(N + M) * D;

    // Output layout (flat concat in return order)
    float* match = (float*)d_out;                 // N x (M+1)
    float* shift = match + (size_t)N * ldc;       // scalar
    float* ious  = shift + 1;                     // N x (M+1)
    float* dxp   = ious + (size_t)N * ldc;        // N x (M+1)
    float* dyp   = dxp  + (size_t)N * ldc;        // N x (M+1)

    dim3 blk(256);

    // GEMM1a: HH[0:N]   = relu(x  @ W0 + b0)
    { dim3 g(D / BN, (N + BM - 1) / BM);
      gemm_wmma<<<g, blk, 0, stream>>>(x,  W0, b0, HH, N, D, Fin, Fin, D, D, 0, 1); }
    // GEMM1b: HH[N:N+M] = relu(rx @ W0 + b0)
    { dim3 g(D / BN, (M + BM - 1) / BM);
      gemm_wmma<<<g, blk, 0, stream>>>(rx, W0, b0, HH + (size_t)N * D, M, D, Fin, Fin, D, D, 0, 1); }
    // GEMM2: G = HH @ W1 + b1
    { dim3 g(D / BN, (N + M + BM - 1) / BM);
      gemm_wmma<<<g, blk, 0, stream>>>(HH, W1, b1, G, N + M, D, D, D, D, D, 0, 0); }
    // GEMM3: match[:, 1:] = G_top @ G_bot^T
    { dim3 g((M + BN - 1) / BN, (N + BM - 1) / BM);
      gemm_wmma<<<g, blk, 0, stream>>>(G, G + (size_t)N * D, nullptr, match + 1,
                                       N, M, D, D, D, ldc, 1, 0); }
    // IoU + distances (columns 1..M)
    { dim3 g((M + 255) / 256, N);
      iou_kernel<<<g, blk, 0, stream>>>(bboxes, rbboxes, ious, dxp, dyp, N, M, ldc); }
    // Zero column 0 of all 2D outputs, shift scalar = 0
    { dim3 g((N + 255) / 256);
      edge_kernel<<<g, blk, 0, stream>>>(match, shift, ious, dxp, dyp, N, ldc); }
}